// WD1d_20675972563801
// MI455X (gfx1250) — compile-verified
//
#include <hip/hip_runtime.h>
#include <hip/hip_bf16.h>
#include <cstdint>

// Problem constants (reference shapes are fixed: x,y = (3000, 800, 8) f32)
#define NT      3000          // time samples per trace
#define KTR     2999          // number of trapezoids / cdf length (NT-1)
#define TRACES  6400          // 800*8 independent traces
#define CH      128           // trapezoids per chunk
#define NCH     24            // ceil(2999/128)
#define TB      256           // block size
#define TROWS   8             // rows per async LDS tile
#define NTILES  17            // ceil((CH+1)/TROWS) -> 136 rows issued, extras clamped
// sum_{i=1..KTR} i = 2999*3000/2
#define SUMW    4498500.0f

// ---------------- CDNA5 async global->LDS helpers (inline asm) ----------------
// GLOBAL_LOAD_ASYNC_TO_LDS_B32: VDST = LDS byte address VGPR, VADDR = 64-bit
// global address. Tracked by ASYNCcnt (CDNA5 ISA).
__device__ __forceinline__ void async_ld_b32(uint32_t lds_byte, const float* g) {
    asm volatile("global_load_async_to_lds_b32 %0, %1, off"
                 :: "v"(lds_byte), "v"((unsigned long long)(uintptr_t)g)
                 : "memory");
}
__device__ __forceinline__ void wait_async_le16() {
    asm volatile("s_wait_asynccnt 0x10" ::: "memory");
}
__device__ __forceinline__ void wait_async_0() {
    asm volatile("s_wait_asynccnt 0x0" ::: "memory");
}
// Generic address of a __shared__ object: high 32 bits = LDS aperture,
// low 32 bits = LDS byte offset (ISA §10.2 aperture calc) -> truncate.
__device__ __forceinline__ uint32_t lds_off(const void* p) {
    return (uint32_t)(uintptr_t)p;
}

// ------------------------------------------------------------------
// Pass A: per (trace, chunk) compute joint min, raw trapezoid sum, and
// weighted sum  Σ trap[k]*(KTR-k)  (== contribution to Σ_i cdfraw[i]).
// Data streamed through LDS with double-buffered async copies; each thread
// consumes only its own column, so no barriers are required — only
// s_wait_asynccnt per tile.
// ------------------------------------------------------------------
__global__ void __launch_bounds__(TB)
wd_passA(const float* __restrict__ x, const float* __restrict__ y,
         float* __restrict__ cMin, float* __restrict__ cTX, float* __restrict__ cWX,
         float* __restrict__ cTY, float* __restrict__ cWY) {
    __shared__ float sx[2][TROWS][TB];
    __shared__ float sy[2][TROWS][TB];

    const int tid  = threadIdx.x;
    const int n    = blockIdx.x * TB + tid;      // trace
    const int c    = blockIdx.y;                 // chunk
    const int k0   = c * CH;
    const int kend = (k0 + CH < KTR) ? (k0 + CH) : KTR;

    auto issue = [&](int tt, int b) {
        const int ebase = k0 + tt * TROWS;
        #pragma unroll
        for (int r = 0; r < TROWS; ++r) {
            int e = ebase + r; if (e > NT - 1) e = NT - 1;   // clamp: harmless dup
            async_ld_b32(lds_off(&sx[b][r][tid]), x + (size_t)e * TRACES + n);
            async_ld_b32(lds_off(&sy[b][r][tid]), y + (size_t)e * TRACES + n);
        }
    };

    issue(0, 0);

    float mn = 3.0e38f;
    float tx = 0.f, wx = 0.f, ty = 0.f, wy = 0.f;
    float pvx = 0.f, pvy = 0.f;

    for (int tt = 0; tt < NTILES; ++tt) {
        if (tt + 1 < NTILES) { issue(tt + 1, (tt + 1) & 1); wait_async_le16(); }
        else                 { wait_async_0(); }
        const int b = tt & 1;
        #pragma unroll
        for (int r = 0; r < TROWS; ++r) {
            const int e = k0 + tt * TROWS + r;
            const float vx = sx[b][r][tid];
            const float vy = sy[b][r][tid];
            mn = fminf(mn, fminf(vx, vy));      // min over extra rows is harmless (global min)
            if (e > k0) {
                const int k = e - 1;
                if (k < kend) {
                    const float w   = (float)(KTR - k);
                    const float trx = 0.5f * (pvx + vx);
                    const float try_ = 0.5f * (pvy + vy);
                    tx += trx;  wx += trx * w;
                    ty += try_; wy += try_ * w;
                }
            }
            pvx = vx; pvy = vy;
        }
    }
    const size_t o = (size_t)c * TRACES + n;
    cMin[o] = mn; cTX[o] = tx; cWX[o] = wx; cTY[o] = ty; cWY[o] = wy;
}

// ------------------------------------------------------------------
// Pass A2: per trace, combine chunks -> joint min, chunk-boundary raw-cdf
// prefixes, and 1/S' where S' = Sraw - min*SUMW  (shift applied analytically).
// ------------------------------------------------------------------
__global__ void __launch_bounds__(TB)
wd_passA2(const float* __restrict__ cMin, const float* __restrict__ cTX,
          const float* __restrict__ cWX, const float* __restrict__ cTY,
          const float* __restrict__ cWY,
          float* __restrict__ mindA, float* __restrict__ invSX, float* __restrict__ invSY,
          float* __restrict__ prefX, float* __restrict__ prefY) {
    const int n = blockIdx.x * blockDim.x + threadIdx.x;
    if (n >= TRACES) return;
    float mn = 3.0e38f, swx = 0.f, swy = 0.f, runx = 0.f, runy = 0.f;
    for (int c = 0; c < NCH; ++c) {
        const size_t o = (size_t)c * TRACES + n;
        prefX[o] = runx; prefY[o] = runy;          // raw cdf just before chunk start
        mn  = fminf(mn, cMin[o]);
        swx += cWX[o]; swy += cWY[o];
        runx += cTX[o]; runy += cTY[o];
    }
    mindA[n] = mn;
    invSX[n] = 1.0f / (swx - mn * SUMW);
    invSY[n] = 1.0f / (swy - mn * SUMW);
}

// ------------------------------------------------------------------
// Pass B: per (trace, i-chunk) monotone two-pointer merge.
//   syn_norm[i] = (cdfrawX[i] - m*(i+1)) * invSX   (non-decreasing)
//   idx(i)      = #{ j : obs_norm[j] < syn_norm[i] }   (searchsorted left)
// Obs pointer is repositioned with a chunk-boundary scan (prefY), then both
// streams advance linearly. acc += ((i+1)-idx)^2 * syn_norm[i].
// ------------------------------------------------------------------
__global__ void __launch_bounds__(TB)
wd_passB(const float* __restrict__ x, const float* __restrict__ y,
         const float* __restrict__ mindA, const float* __restrict__ invSXA,
         const float* __restrict__ invSYA,
         const float* __restrict__ prefX, const float* __restrict__ prefY,
         float* __restrict__ partial) {
    const int tid = threadIdx.x;
    const int n   = blockIdx.x * TB + tid;
    const int c   = blockIdx.y;
    const int k0  = c * CH;
    const int iend = (k0 + CH < KTR) ? (k0 + CH) : KTR;

    const float m   = mindA[n];
    const float isx = invSXA[n];
    const float isy = invSYA[n];

    // syn stream state
    float cx = prefX[(size_t)c * TRACES + n];
    float px = x[(size_t)k0 * TRACES + n];

    // v at i=k0 (identical ops to the loop body -> bitwise consistent)
    {
        const float nx0 = x[(size_t)(k0 + 1) * TRACES + n];
        const float cx0 = cx + 0.5f * (px + nx0);
        const float v0  = (cx0 - m * (float)(k0 + 1)) * isx;

        // boundary scan: B(t) = obs_norm[t*CH - 1] from stored prefixes
        int cj = 0;
        for (int t = 1; t < NCH; ++t) {
            const float B = (prefY[(size_t)t * TRACES + n] - m * (float)(t * CH)) * isy;
            if (B < v0) cj = t; else break;
        }
        // init obs stream at candidate jj = cj*CH
        int jj = cj * CH;
        float cy = prefY[(size_t)cj * TRACES + n];
        float py = y[(size_t)jj * TRACES + n];
        {
            const float ny = y[(size_t)(jj + 1) * TRACES + n];
            cy += 0.5f * (py + ny); py = ny;
        }
        float nxt = (cy - m * (float)(jj + 1)) * isy;   // obs_norm[jj]

        float acc = 0.0f;
        for (int i = k0; i < iend; ++i) {
            const float nx = x[(size_t)(i + 1) * TRACES + n];
            cx += 0.5f * (px + nx); px = nx;
            const float v = (cx - m * (float)(i + 1)) * isx;
            while (jj < KTR && nxt < v) {
                ++jj;
                if (jj < KTR) {
                    const float ny = y[(size_t)(jj + 1) * TRACES + n];
                    cy += 0.5f * (py + ny); py = ny;
                    nxt = (cy - m * (float)(jj + 1)) * isy;
                } else {
                    nxt = 3.0e38f;
                }
            }
            // idx = jj (<= KTR == nt-1, so the reference clamp is a no-op)
            const float diff = (float)(i + 1) - (float)jj;
            acc += diff * diff * v;
        }
        partial[(size_t)c * TRACES + n] = acc;
    }
}

// ------------------------------------------------------------------
// Deterministic final reduction (single block, fixed order; no fp atomics).
// ------------------------------------------------------------------
__global__ void __launch_bounds__(TB)
wd_reduce(const float* __restrict__ partial, float* __restrict__ out, int mtot) {
    __shared__ float s[TB];
    float a = 0.f;
    for (int i = threadIdx.x; i < mtot; i += TB) a += partial[i];
    s[threadIdx.x] = a;
    __syncthreads();
    for (int w = TB / 2; w > 0; w >>= 1) {
        if (threadIdx.x < w) s[threadIdx.x] += s[threadIdx.x + w];
        __syncthreads();
    }
    if (threadIdx.x == 0) out[0] = s[0];
}

extern "C" void kernel_launch(void* const* d_in, const int* in_sizes, int n_in,
                              void* d_out, int out_size, void* d_ws, size_t ws_size,
                              hipStream_t stream) {
    const float* x = (const float*)d_in[0];
    const float* y = (const float*)d_in[1];
    float* out = (float*)d_out;

    // Workspace layout (floats). Total = 8*M + 3*TRACES ~ 5.0 MB.
    const size_t M = (size_t)TRACES * NCH;   // 153600
    float* ws    = (float*)d_ws;
    float* cMin  = ws + 0 * M;
    float* cTX   = ws + 1 * M;
    float* cWX   = ws + 2 * M;
    float* cTY   = ws + 3 * M;
    float* cWY   = ws + 4 * M;
    float* prefX = ws + 5 * M;
    float* prefY = ws + 6 * M;
    float* part  = ws + 7 * M;
    float* mind  = ws + 8 * M;
    float* invSX = mind + TRACES;
    float* invSY = mind + 2 * TRACES;

    dim3 blk(TB);
    dim3 grdA(TRACES / TB, NCH);   // (25, 24)
    wd_passA<<<grdA, blk, 0, stream>>>(x, y, cMin, cTX, cWX, cTY, cWY);

    dim3 grd2(TRACES / TB);        // 25
    wd_passA2<<<grd2, blk, 0, stream>>>(cMin, cTX, cWX, cTY, cWY,
                                        mind, invSX, invSY, prefX, prefY);

    wd_passB<<<grdA, blk, 0, stream>>>(x, y, mind, invSX, invSY, prefX, prefY, part);

    wd_reduce<<<dim3(1), blk, 0, stream>>>(part, out, (int)M);
}